// SolveScsQP_33938831573159
// MI455X (gfx1250) — compile-verified
//
#include <hip/hip_runtime.h>

// Batched ADMM QP solver for MI455X (gfx1250, wave32).
// One workgroup (160 threads = 5 waves) per batch; all per-batch matrices live
// in LDS for the whole 800-iteration loop. Setup GEMMs (A^T A, Kinv*A^T,
// P = Amat*R) run on V_WMMA_F32_16X16X4_F32; the sequential GEMV iterations run
// on VALU FMA with conflict-free float2 LDS reads (row stride 146 floats).

typedef float v2f __attribute__((ext_vector_type(2)));
typedef float v8f __attribute__((ext_vector_type(8)));

namespace {
constexpr int   kNX     = 64;
constexpr int   kNEQ    = 16;
constexpr int   kNIN    = 64;
constexpr int   kM      = kNEQ + kNIN;   // 80 stacked constraints
constexpr int   kNV     = kNX + kM;      // 144 = rows of fused matrix [R; P]
constexpr int   kIters  = 800;
constexpr float kSigma  = 1e-6f;         // RHO == 1.0 exactly, folded away
constexpr int   kThreads = 160;          // 5 wave32 waves

// LDS layout (float offsets). kMStride=146: 18*lane mod 64 distinct for all
// 32 lanes -> conflict-free strided float2 GEMV reads.
constexpr int kMStride   = 146;
constexpr int kAugStride = 130;
constexpr int kAmStride  = 66;
constexpr int offM   = 0;
constexpr int offAug = offM   + kNV * kMStride;    // 21024
constexpr int offAm  = offAug + kNX * kAugStride;  // +8320
constexpr int offV   = offAm  + kM  * kAmStride;   // +5280  (even -> f2 aligned)
constexpr int offCl  = offV   + kMStride;
constexpr int offPl  = offCl  + kNX;
constexpr int offFv  = offPl  + kNX;
constexpr int kLdsFloats = offFv + kNX;            // 34962 floats ~ 137 KB
}  // namespace

__global__ __launch_bounds__(kThreads)
void admm_qp_kernel(const float* __restrict__ Q, const float* __restrict__ p,
                    const float* __restrict__ Aeq, const float* __restrict__ beq,
                    const float* __restrict__ G, const float* __restrict__ h,
                    float* __restrict__ out) {
  extern __shared__ float lds[];
  float* Mlds = lds + offM;    // [144][146]  fused [R; P]
  float* Aug  = lds + offAug;  // [64][130]   [K | I] -> [I | Kinv]
  float* Am   = lds + offAm;   // [80][66]    Amat = [A; G]
  float* vv   = lds + offV;    // [144]       v = [x; z - y]
  float* cl   = lds + offCl;   // [64]        c = Kinv p
  float* pl   = lds + offPl;   // [64]
  float* fv   = lds + offFv;   // [64]        GJ elimination factors

  const int b    = blockIdx.x;
  const int t    = threadIdx.x;
  const int wave = t >> 5;
  const int lane = t & 31;
  const int frag   = lane & 15;          // fragment M/N index
  const int kpb    = (lane >> 4) << 1;   // A/B fragment K pair base (0 or 2)
  const int rowoff = (lane >> 4) << 3;   // C/D fragment row offset (0 or 8)

  // ---- Stage global data into LDS ----------------------------------------
  const float* Ab = Aeq + (size_t)b * kNEQ * kNX;
  const float* Gb = G   + (size_t)b * kNIN * kNX;
  const float* Qb = Q   + (size_t)b * kNX * kNX;
  for (int i = t; i < kNEQ * kNX; i += kThreads)
    Am[(i >> 6) * kAmStride + (i & 63)] = Ab[i];
  for (int i = t; i < kNIN * kNX; i += kThreads)
    Am[(kNEQ + (i >> 6)) * kAmStride + (i & 63)] = Gb[i];
  for (int i = t; i < kNX * kNX; i += kThreads) {
    int r = i >> 6, c = i & 63;
    float q = Qb[i];
    if (r == c) q += kSigma;
    Aug[r * kAugStride + c] = q;                         // K (so far Q + sigma I)
    Aug[r * kAugStride + kNX + c] = (r == c) ? 1.f : 0.f; // identity
  }
  for (int i = t; i < kNX; i += kThreads) pl[i] = p[(size_t)b * kNX + i];
  __syncthreads();

  // ---- K += Am^T * Am   (WMMA f32 16x16x4, M=N=64, K=80; 16 tiles / 5 waves)
  for (int tile = wave; tile < 16; tile += 5) {
    const int i0 = (tile >> 2) << 4, j0 = (tile & 3) << 4;
    v8f acc;
#pragma unroll
    for (int g = 0; g < 8; ++g)
      acc[g] = Aug[(i0 + g + rowoff) * kAugStride + j0 + frag];
#pragma unroll
    for (int k = 0; k < kM; k += 4) {
      v2f a, bb;  // A[m][k] = Am[k][i0+m], B[k][n] = Am[k][j0+n]
      a.x  = Am[(k + kpb)     * kAmStride + i0 + frag];
      a.y  = Am[(k + kpb + 1) * kAmStride + i0 + frag];
      bb.x = Am[(k + kpb)     * kAmStride + j0 + frag];
      bb.y = Am[(k + kpb + 1) * kAmStride + j0 + frag];
      acc = __builtin_amdgcn_wmma_f32_16x16x4_f32(false, a, false, bb,
                                                  (short)0, acc, false, false);
    }
#pragma unroll
    for (int g = 0; g < 8; ++g)
      Aug[(i0 + g + rowoff) * kAugStride + j0 + frag] = acc[g];
  }

  // ---- Gauss-Jordan: [K | I] -> [I | Kinv] (pivotless: K is SPD) ----------
  for (int k = 0; k < kNX; ++k) {
    __syncthreads();  // also covers the WMMA stores at k==0
    const float pinv = 1.0f / Aug[k * kAugStride + k];
    if (t < kNX) fv[t] = Aug[t * kAugStride + k];
    __syncthreads();
    for (int c = t; c < 2 * kNX; c += kThreads) Aug[k * kAugStride + c] *= pinv;
    __syncthreads();
    for (int idx = t; idx < kNX * 2 * kNX; idx += kThreads) {
      int r = idx >> 7, c = idx & 127;
      if (r != k)
        Aug[r * kAugStride + c] =
            fmaf(-fv[r], Aug[k * kAugStride + c], Aug[r * kAugStride + c]);
    }
  }
  __syncthreads();

  // ---- R-left = sigma * Kinv ;  c = Kinv * p ------------------------------
  for (int idx = t; idx < kNX * kNX; idx += kThreads) {
    int r = idx >> 6, c = idx & 63;
    Mlds[r * kMStride + c] = kSigma * Aug[r * kAugStride + kNX + c];
  }
  if (t < kNX) {
    float s = 0.f;
    for (int k = 0; k < kNX; ++k)
      s = fmaf(Aug[t * kAugStride + kNX + k], pl[k], s);
    cl[t] = s;
  }
  // ---- R-right = Kinv * Am^T (WMMA, M=64, N=80, K=64; 20 tiles) -----------
  for (int tile = wave; tile < 20; tile += 5) {
    const int i0 = (tile / 5) << 4, j0 = (tile % 5) << 4;
    v8f acc;
#pragma unroll
    for (int g = 0; g < 8; ++g) acc[g] = 0.f;
#pragma unroll
    for (int k = 0; k < kNX; k += 4) {
      v2f a, bb;  // A[m][k]=Kinv[i0+m][k], B[k][n]=Am[j0+n][k]
      a.x  = Aug[(i0 + frag) * kAugStride + kNX + k + kpb];
      a.y  = Aug[(i0 + frag) * kAugStride + kNX + k + kpb + 1];
      bb.x = Am[(j0 + frag) * kAmStride + k + kpb];
      bb.y = Am[(j0 + frag) * kAmStride + k + kpb + 1];
      acc = __builtin_amdgcn_wmma_f32_16x16x4_f32(false, a, false, bb,
                                                  (short)0, acc, false, false);
    }
#pragma unroll
    for (int g = 0; g < 8; ++g)
      Mlds[(i0 + g + rowoff) * kMStride + kNX + j0 + frag] = acc[g];
  }
  __syncthreads();

  // ---- P = Am * R into M rows 64..143 (WMMA, M=80, N=144, K=64; 45 tiles) -
  for (int tile = wave; tile < 45; tile += 5) {
    const int i0 = (tile / 9) << 4, j0 = (tile % 9) << 4;
    v8f acc;
#pragma unroll
    for (int g = 0; g < 8; ++g) acc[g] = 0.f;
#pragma unroll
    for (int k = 0; k < kNX; k += 4) {
      v2f a, bb;  // A[m][k]=Am[i0+m][k], B[k][n]=R[k][j0+n]
      a.x  = Am[(i0 + frag) * kAmStride + k + kpb];
      a.y  = Am[(i0 + frag) * kAmStride + k + kpb + 1];
      bb.x = Mlds[(k + kpb)     * kMStride + j0 + frag];
      bb.y = Mlds[(k + kpb + 1) * kMStride + j0 + frag];
      acc = __builtin_amdgcn_wmma_f32_16x16x4_f32(false, a, false, bb,
                                                  (short)0, acc, false, false);
    }
#pragma unroll
    for (int g = 0; g < 8; ++g)
      Mlds[(kNX + i0 + g + rowoff) * kMStride + j0 + frag] = acc[g];
  }

  // ---- per-thread constants: cd = [c; d] with d = Am*c; bounds; y ---------
  float cdreg = 0.f, yreg = 0.f, lreg = 0.f, ureg = 0.f;
  if (t < kNX) {
    cdreg = cl[t];
  } else if (t < kNV) {
    const int j = t - kNX;
    float s = 0.f;
    for (int k = 0; k < kNX; ++k) s = fmaf(Am[j * kAmStride + k], cl[k], s);
    cdreg = s;
    if (j < kNEQ) { lreg = ureg = beq[(size_t)b * kNEQ + j]; }
    else          { lreg = -1e8f; ureg = h[(size_t)b * kNIN + (j - kNEQ)]; }
  }
  if (t < kNV) vv[t] = 0.f;  // x0 = 0, z0 - y0 = 0
  __syncthreads();

  // ---- 800 ADMM iterations: [x; Ax] = [R;P] v - [c;d]; clip; dual update --
  const float2* vrow = (const float2*)vv;
  const float2* mrow = (const float2*)(Mlds + (size_t)(t < kNV ? t : 0) * kMStride);
  float xout = 0.f;
  for (int it = 0; it < kIters; ++it) {
    float acc = -cdreg;
    if (t < kNV) {
#pragma unroll 8
      for (int kk = 0; kk < kNV / 2; ++kk) {
        const float2 m2 = mrow[kk];
        const float2 v2 = vrow[kk];
        acc = fmaf(m2.x, v2.x, acc);
        acc = fmaf(m2.y, v2.y, acc);
      }
    }
    __syncthreads();                 // all reads of vv complete
    if (t < kNX) {
      vv[t] = acc;                   // x_{i+1}
      xout = acc;
    } else if (t < kNV) {
      const float ax = acc;          // (Amat x_{i+1})_j
      float z = fminf(fmaxf(ax + yreg, lreg), ureg);
      yreg += ax - z;
      vv[t] = z - yreg;              // w = z_{i+1} - y_{i+1}
    }
    __syncthreads();                 // vv ready for next iteration
  }
  if (t < kNX) out[(size_t)b * kNX + t] = xout;
}

extern "C" void kernel_launch(void* const* d_in, const int* in_sizes, int n_in,
                              void* d_out, int out_size, void* d_ws, size_t ws_size,
                              hipStream_t stream) {
  (void)n_in; (void)out_size; (void)d_ws; (void)ws_size;
  const float* Q   = (const float*)d_in[0];
  const float* p   = (const float*)d_in[1];
  const float* A   = (const float*)d_in[2];
  const float* bev = (const float*)d_in[3];
  const float* G   = (const float*)d_in[4];
  const float* h   = (const float*)d_in[5];
  float* out = (float*)d_out;

  const int B = in_sizes[0] / (kNX * kNX);
  const size_t shmem = (size_t)kLdsFloats * sizeof(float);
  admm_qp_kernel<<<dim3(B), dim3(kThreads), shmem, stream>>>(Q, p, A, bev, G, h, out);
}